// HierarchicalDistanceLoss_30966714204286
// MI455X (gfx1250) — compile-verified
//
#include <hip/hip_runtime.h>

// ---------------------------------------------------------------------------
// HierarchicalDistanceLoss for MI455X (gfx1250, wave32).
//
// Roofline: 160MB logits + 4MB labels read, 4MB factor written => ~7.2us at
// 23.3 TB/s HBM (working set < 192MB L2, so timed replays may run from L2).
// Memory-bound => float4 (global_load_b128) row loads, default RT cache hints.
//
// VALU diet vs round 1 (budget ~3.5 lane-ops/byte => ~590 ops/row):
//  - x[label] fetched with one global_load_b32 (L0 hit; row just loaded)
//    instead of a 40-deep cmp/cndmask ladder.
//  - raw v_exp_f32/v_log_f32 (base-2 hardware transcendentals, TRANS pipe
//    co-executes with VALU) instead of guarded __expf/__logf.
//
// CDNA5-specific path: per-wave reduction of the 64 ce*factor partials
// (2 per lane) is one V_WMMA_F32_16X16X4_F32 against an all-ones B matrix:
// A (16x4 f32) is exactly 2 VGPRs/lane, D rows hold A row-sums, a single
// shfl_xor(16) completes the wave sum. EXEC is all-ones at that point.
// ---------------------------------------------------------------------------

typedef __attribute__((ext_vector_type(2))) float v2f;
typedef __attribute__((ext_vector_type(8))) float v8f;

#define NCLS 40
#define TPB 256
#define ROWS_PER_THREAD 2

#define LOG2E 1.44269504088896340736f
#define LN2   0.69314718055994530942f

__device__ __forceinline__ float fast_exp2(float x) {
#if __has_builtin(__builtin_amdgcn_exp2f)
    return __builtin_amdgcn_exp2f(x);     // raw v_exp_f32
#else
    return exp2f(x);
#endif
}
__device__ __forceinline__ float fast_log2(float x) {
#if __has_builtin(__builtin_amdgcn_logf)
    return __builtin_amdgcn_logf(x);      // raw v_log_f32
#else
    return log2f(x);
#endif
}

__global__ __launch_bounds__(TPB) void hdl_main(
    const float* __restrict__ logits,   // [B, 40]
    const int*   __restrict__ labels,   // [B]
    const float* __restrict__ dis,      // [40, 40]
    float*       __restrict__ out,      // out[0]=loss (kernel2), out+1 = factor[B]
    float*       __restrict__ partials, // [gridDim.x] block partial sums
    int B)
{
    __shared__ float sdis[NCLS * NCLS];
    __shared__ float sred[TPB];
    __shared__ float swave[TPB / 32];

    const int tid = threadIdx.x;

    // Stage dis_matrix into LDS and compute dis_max (block-redundant; 6.4KB L2-resident).
    float lmax = -3.402823466e38f;
    for (int i = tid; i < NCLS * NCLS; i += TPB) {
        float v = dis[i];
        sdis[i] = v;
        lmax = fmaxf(lmax, v);
    }
    sred[tid] = lmax;
    __syncthreads();
    #pragma unroll
    for (int s = TPB / 2; s > 0; s >>= 1) {
        if (tid < s) sred[tid] = fmaxf(sred[tid], sred[tid + s]);
        __syncthreads();
    }
    const float dmax = sred[0];

    // Each thread processes ROWS_PER_THREAD consecutive rows.
    float accs[ROWS_PER_THREAD];
    const long long gbase =
        ((long long)blockIdx.x * TPB + tid) * ROWS_PER_THREAD;

    #pragma unroll
    for (int rr = 0; rr < ROWS_PER_THREAD; ++rr) {
        const long long r = gbase + rr;
        float acc = 0.0f;
        if (r < (long long)B) {
            const float* rowp = logits + r * (long long)NCLS;
            // 40 floats = 10 x float4; 160B rows stay 16B-aligned.
            const float4* rp = reinterpret_cast<const float4*>(rowp);
            float x[NCLS];
            #pragma unroll
            for (int k = 0; k < NCLS / 4; ++k) {
                float4 q = rp[k];
                x[4 * k + 0] = q.x; x[4 * k + 1] = q.y;
                x[4 * k + 2] = q.z; x[4 * k + 3] = q.w;
            }
            const int lab = labels[r];
            // Direct gather of x[label]; hits L0 (row just loaded). Replaces
            // a 40-deep cmp/cndmask ladder.
            const float xl = rowp[lab];

            // argmax, first occurrence (strict >) matches jnp.argmax.
            float m = x[0];
            int   am = 0;
            #pragma unroll
            for (int j = 1; j < NCLS; ++j) {
                if (x[j] > m) { m = x[j]; am = j; }
            }

            // sum exp(x-m) = sum exp2((x-m)*log2e); per term: fma + v_exp + add.
            const float mb = -m * LOG2E;
            float s = 0.0f;
            #pragma unroll
            for (int j = 0; j < NCLS; ++j) {
                s += fast_exp2(fmaf(x[j], LOG2E, mb));
            }
            // ce = m + ln(s) - x[lab];  ln(s) = log2(s)*ln2;  s in [1,40].
            const float ce = fmaf(fast_log2(s), LN2, m - xl);

            const bool gnr = (lab == 0);
            const bool pnr = (am == 0);
            float factor;
            if (gnr && !pnr)      factor = 2.0f * dmax;
            else if (pnr && !gnr) factor = dmax;
            else                  factor = sdis[lab * NCLS + am];

            out[1 + r] = factor;
            acc = ce * factor;
        }
        accs[rr] = acc;
    }

    // ---- Wave32 sum of the 64 per-lane partials via V_WMMA_F32_16X16X4_F32 ----
    // A (16x4 f32) = 2 VGPRs/lane: our 64 partials exactly fill it.
    // Lanes 0-15 see rows 0-7 in d[0..7], lanes 16-31 see rows 8-15:
    // wave total = sum(d) + shfl_xor(sum(d), 16).
    v2f a;    a.x = accs[0];  a.y = accs[1];
    v2f ones; ones.x = 1.0f;  ones.y = 1.0f;
    v8f cz = {};
    v8f d = __builtin_amdgcn_wmma_f32_16x16x4_f32(
        /*neg_a=*/false, a, /*neg_b=*/false, ones,
        /*c_mod=*/(short)0, cz, /*reuse_a=*/false, /*reuse_b=*/false);
    float t = d[0] + d[1] + d[2] + d[3] + d[4] + d[5] + d[6] + d[7];
    t += __shfl_xor(t, 16, 32);

    const int lane = tid & 31;
    const int wid  = tid >> 5;
    if (lane == 0) swave[wid] = t;
    __syncthreads();
    if (tid == 0) {
        float bt = 0.0f;
        #pragma unroll
        for (int w = 0; w < TPB / 32; ++w) bt += swave[w];
        partials[blockIdx.x] = bt;
    }
}

// Deterministic final reduction (no float atomics => bit-identical replays).
__global__ __launch_bounds__(TPB) void hdl_reduce(
    const float* __restrict__ partials, int n,
    float* __restrict__ out, float invB)
{
    __shared__ float sred[TPB];
    float acc = 0.0f;
    for (int i = threadIdx.x; i < n; i += TPB) acc += partials[i];
    sred[threadIdx.x] = acc;
    __syncthreads();
    #pragma unroll
    for (int s = TPB / 2; s > 0; s >>= 1) {
        if (threadIdx.x < s) sred[threadIdx.x] += sred[threadIdx.x + s];
        __syncthreads();
    }
    if (threadIdx.x == 0) out[0] = sred[0] * invB;
}

extern "C" void kernel_launch(void* const* d_in, const int* in_sizes, int n_in,
                              void* d_out, int out_size, void* d_ws, size_t ws_size,
                              hipStream_t stream)
{
    const float* logits = (const float*)d_in[0];   // [B,40] f32
    const int*   labels = (const int*)d_in[1];     // [B] (JAX int64 -> int32 without x64)
    const float* dis    = (const float*)d_in[2];   // [40,40] f32
    // d_in[3] = lca_matrix: unused by the reference.

    float* out = (float*)d_out;   // [0]=loss, [1..B]=factor
    float* ws  = (float*)d_ws;    // block partials

    const int B = in_sizes[1];
    const int rows_per_block = TPB * ROWS_PER_THREAD;
    const int grid = (B + rows_per_block - 1) / rows_per_block;

    hdl_main<<<grid, TPB, 0, stream>>>(logits, labels, dis, out, ws, B);
    hdl_reduce<<<1, TPB, 0, stream>>>(ws, grid, out, 1.0f / (float)B);
}